// TokenMergeModule_9070970929186
// MI455X (gfx1250) — compile-verified
//
#include <hip/hip_runtime.h>
#include <stdint.h>

#define B_ 8
#define S_ 4096
#define D_ 512
#define N_ 4096
#define G_ 64
#define R_ 1024
#define SK_ 3072          // S - R kept tokens
#define EPS_MERGE 1e-8f
#define EPS_NORM  1e-12f

typedef float v2f __attribute__((ext_vector_type(2)));
typedef float v8f __attribute__((ext_vector_type(8)));

// ---------------------------------------------------------------------------
// g[b,s,n] = sum_d x[b,s,d] * W[n,d]   via V_WMMA_F32_16X16X4_F32 (wave32)
// Block = 128 threads (4 waves). Each block: 16 rows of x, each wave: 16 of
// the 64 output columns. K loop over D in steps of 4.
// A frag (16x4 f32): lane L -> M = L&15, holds K = 2*(L>>4) + {0,1}
// B frag (4x16 f32): lane L -> N = L&15, holds K = 2*(L>>4) + {0,1}
// D frag (16x16 f32): VGPR j -> M = j + 8*(L>>4), N = L&15
// ---------------------------------------------------------------------------
__global__ __launch_bounds__(128)
void k_gemm_wmma(const float* __restrict__ x, const float* __restrict__ W,
                 float* __restrict__ g) {
  const int blk  = blockIdx.x;              // B * (S/16)
  const int b    = blk / (S_ / 16);
  const int s0   = (blk % (S_ / 16)) * 16;
  const int wave = threadIdx.x >> 5;        // 0..3 -> N tile
  const int lane = threadIdx.x & 31;
  const int m    = lane & 15;
  const int kh   = lane >> 4;               // 0/1 -> K offset 0/2
  const int n    = wave * 16 + m;

  const float* xrow = x + (size_t)(b * S_ + s0 + m) * D_ + 2 * kh;
  const float* wrow = W + (size_t)n * D_ + 2 * kh;

  v8f acc = {0.f, 0.f, 0.f, 0.f, 0.f, 0.f, 0.f, 0.f};
#pragma unroll 4
  for (int k = 0; k < D_; k += 4) {
    v2f a  = *(const v2f*)(xrow + k);
    v2f bf = *(const v2f*)(wrow + k);
    acc = __builtin_amdgcn_wmma_f32_16x16x4_f32(
        /*neg_a=*/false, a, /*neg_b=*/false, bf,
        /*c_mod=*/(short)0, acc, /*reuse_a=*/false, /*reuse_b=*/false);
  }

  float* gout = g + (size_t)(b * S_ + s0) * G_ + wave * 16 + m;
#pragma unroll
  for (int j = 0; j < 8; ++j) {
    int mm = j + 8 * kh;
    gout[(size_t)mm * G_] = acc[j];
  }
}

// ---------------------------------------------------------------------------
// Row norms of g  (one thread per row, 16x float4)
// ---------------------------------------------------------------------------
__global__ void k_norms(const float* __restrict__ g, float* __restrict__ gn) {
  int t = blockIdx.x * blockDim.x + threadIdx.x;  // B*S
  if (t >= B_ * S_) return;
  const float4* row = (const float4*)(g + (size_t)t * G_);
  float s = 0.f;
#pragma unroll
  for (int i = 0; i < G_ / 4; ++i) {
    float4 v = row[i];
    s += v.x * v.x + v.y * v.y + v.z * v.z + v.w * v.w;
  }
  gn[t] = sqrtf(s);
}

// ---------------------------------------------------------------------------
// Adjacent cosine similarity: sim[b,s] = <g_s, g_{s+1}> / (max(n0,e)*max(n1,e))
// stride S per batch; s = S-1 unused.
// ---------------------------------------------------------------------------
__global__ void k_sim(const float* __restrict__ g, const float* __restrict__ gn,
                      float* __restrict__ sim) {
  int t = blockIdx.x * blockDim.x + threadIdx.x;  // B*S
  if (t >= B_ * S_) return;
  int s = t & (S_ - 1);
  if (s == S_ - 1) return;
  const float4* r0 = (const float4*)(g + (size_t)t * G_);
  const float4* r1 = (const float4*)(g + (size_t)(t + 1) * G_);
  float d = 0.f;
#pragma unroll
  for (int i = 0; i < G_ / 4; ++i) {
    float4 a = r0[i], c = r1[i];
    d += a.x * c.x + a.y * c.y + a.z * c.z + a.w * c.w;
  }
  float n0 = fmaxf(gn[t], EPS_NORM), n1 = fmaxf(gn[t + 1], EPS_NORM);
  sim[t] = d / (n0 * n1);
}

// ---------------------------------------------------------------------------
// Per-batch: stable argsort(-sim) via bitonic sort of 64-bit keys in LDS,
// then serial greedy disjoint pair selection (lane 0), then keep-list build.
// key = (~monotonic(simbits)) << 32 | idx  -> ascending sort == stable
// descending-by-sim order. Edge S-1 padded with max key.
// ---------------------------------------------------------------------------
__global__ __launch_bounds__(512)
void k_select(const float* __restrict__ sim, int* __restrict__ sel,
              int* __restrict__ keep) {
  __shared__ unsigned long long keys[S_];
  __shared__ unsigned char used[S_ + 1];
  __shared__ unsigned char selL[S_];
  const int b = blockIdx.x;
  const int tid = threadIdx.x;

  for (int i = tid; i < S_; i += 512) {
    unsigned long long key;
    if (i < S_ - 1) {
      unsigned u = __float_as_uint(sim[b * S_ + i]);
      u = (u & 0x80000000u) ? ~u : (u | 0x80000000u);  // ascending float map
      unsigned dk = ~u;                                // descending by sim
      key = ((unsigned long long)dk << 32) | (unsigned)i;
    } else {
      key = 0xFFFFFFFFFFFFFFFFull;                     // pad: sorts last
    }
    keys[i] = key;
    used[i] = 0;
    selL[i] = 0;
  }
  if (tid == 0) used[S_] = 0;
  __syncthreads();

  // bitonic sort ascending, 4096 elements, 512 threads
  for (int k = 2; k <= S_; k <<= 1) {
    for (int j = k >> 1; j > 0; j >>= 1) {
      for (int i = tid; i < S_; i += 512) {
        int ixj = i ^ j;
        if (ixj > i) {
          unsigned long long a = keys[i], c = keys[ixj];
          bool up = ((i & k) == 0);
          if ((a > c) == up) { keys[i] = c; keys[ixj] = a; }
        }
      }
      __syncthreads();
    }
  }

  if (tid == 0) {
    int cnt = 0;
    for (int i = 0; i < S_ && cnt < R_; ++i) {
      unsigned p = (unsigned)(keys[i] & 0xFFFFFFFFull);
      if (p >= (unsigned)(S_ - 1)) continue;
      if (!used[p] && !used[p + 1]) {
        used[p] = 1;
        used[p + 1] = 1;
        selL[p] = 1;
        ++cnt;
      }
    }
    // stable compaction: drop[i] = sel[i-1]; keep first SK_ survivors
    int j = 0;
    for (int i = 0; i < S_; ++i) {
      bool drop = (i > 0) && selL[i - 1];
      if (!drop) {
        if (j < SK_) keep[b * SK_ + j] = i;
        ++j;
      }
    }
  }
  __syncthreads();
  for (int p = tid; p < S_; p += 512)
    sel[b * S_ + p] = (p < S_ - 1) ? (int)selL[p] : 0;
}

// ---------------------------------------------------------------------------
// Gather/merge x: one block per output row (512 floats = 128 float4)
// ---------------------------------------------------------------------------
__global__ __launch_bounds__(128)
void k_gather_x(const float* __restrict__ x, const float* __restrict__ gn,
                const int* __restrict__ sel, const int* __restrict__ keep,
                float* __restrict__ xo) {
  int row = blockIdx.x;  // B*SK_
  int b = row / SK_;
  int i = keep[row];
  int se = (i < S_ - 1) ? sel[b * S_ + i] : 0;
  const float4* xi = (const float4*)(x + (size_t)(b * S_ + i) * D_);
  float4* out = (float4*)(xo + (size_t)row * D_);
  int c = threadIdx.x;  // 0..127
  float4 a = xi[c];
  if (se) {
    const float4* xj = (const float4*)(x + (size_t)(b * S_ + i + 1) * D_);
    float w0 = gn[b * S_ + i], w1 = gn[b * S_ + i + 1];
    float inv = 1.f / (w0 + w1 + EPS_MERGE);
    float4 d = xj[c];
    a.x = (w0 * a.x + w1 * d.x) * inv;
    a.y = (w0 * a.y + w1 * d.y) * inv;
    a.z = (w0 * a.z + w1 * d.z) * inv;
    a.w = (w0 * a.w + w1 * d.w) * inv;
  }
  out[c] = a;
}

// ---------------------------------------------------------------------------
// Gather/accumulate source: one block per output row (4096 floats)
// ---------------------------------------------------------------------------
__global__ __launch_bounds__(256)
void k_gather_src(const float* __restrict__ src, const int* __restrict__ sel,
                  const int* __restrict__ keep, float* __restrict__ so) {
  int row = blockIdx.x;  // B*SK_
  int b = row / SK_;
  int i = keep[row];
  int se = (i < S_ - 1) ? sel[b * S_ + i] : 0;
  const float4* s0p = (const float4*)(src + (size_t)(b * S_ + i) * N_);
  const float4* s1p = (const float4*)(src + (size_t)(b * S_ + i + 1) * N_);
  float4* out = (float4*)(so + (size_t)row * N_);
#pragma unroll
  for (int c = threadIdx.x; c < N_ / 4; c += 256) {
    float4 a = s0p[c];
    if (se) {
      float4 d = s1p[c];
      a.x += d.x; a.y += d.y; a.z += d.z; a.w += d.w;
    }
    out[c] = a;
  }
}

__global__ void k_gather_pos(const int* __restrict__ pos,
                             const int* __restrict__ keep,
                             int* __restrict__ po) {
  int t = blockIdx.x * blockDim.x + threadIdx.x;
  if (t >= B_ * SK_) return;
  int b = t / SK_;
  po[t] = pos[b * S_ + keep[t]];
}

// ---------------------------------------------------------------------------
extern "C" void kernel_launch(void* const* d_in, const int* in_sizes, int n_in,
                              void* d_out, int out_size, void* d_ws,
                              size_t ws_size, hipStream_t stream) {
  (void)in_sizes; (void)n_in; (void)out_size; (void)ws_size;
  const float* x   = (const float*)d_in[0];
  const float* src = (const float*)d_in[1];
  const int*   pos = (const int*)d_in[2];
  const float* W   = (const float*)d_in[3];

  char* ws = (char*)d_ws;
  float* g   = (float*)(ws);                       // B*S*G  = 8 MB
  float* gn  = (float*)(ws + 8388608);             // B*S    = 128 KB
  float* sim = (float*)(ws + 8388608 + 131072);    // B*S    = 128 KB
  int*   sel = (int*)  (ws + 8388608 + 262144);    // B*S    = 128 KB
  int*   keep= (int*)  (ws + 8388608 + 393216);    // B*SK   = 96 KB

  float* xo = (float*)d_out;
  float* so = xo + (size_t)B_ * SK_ * D_;
  int*   po = (int*)(so + (size_t)B_ * SK_ * N_);

  k_gemm_wmma<<<B_ * (S_ / 16), 128, 0, stream>>>(x, W, g);
  k_norms<<<(B_ * S_) / 256, 256, 0, stream>>>(g, gn);
  k_sim<<<(B_ * S_) / 256, 256, 0, stream>>>(g, gn, sim);
  k_select<<<B_, 512, 0, stream>>>(sim, sel, keep);
  k_gather_x<<<B_ * SK_, 128, 0, stream>>>(x, gn, sel, keep, xo);
  k_gather_src<<<B_ * SK_, 256, 0, stream>>>(src, sel, keep, so);
  k_gather_pos<<<(B_ * SK_ + 255) / 256, 256, 0, stream>>>(pos, keep, po);
}